// SoftMaxPool2d_50130858279617
// MI455X (gfx1250) — compile-verified
//
#include <hip/hip_runtime.h>
#include <hip/hip_bf16.h>

// ---------------------------------------------------------------------------
// SoftMaxPool2d: in [32,64,112,112] f32, 3x3 window, stride 2, pad 1, T=1.0
// out [32,64,56,56] f32 ;  out = sum(softmax(w)*w) over each 3x3 window.
// Memory-bound (~128 MB traffic -> ~5.5us at 23.3 TB/s). Strategy: stage
// padded input tiles in LDS via gfx1250 async global->LDS copies, compute
// from LDS (stride-2 reads are bank-conflict-free on 64 banks / wave32).
// ---------------------------------------------------------------------------

#if defined(__has_builtin)
#  if __has_builtin(__builtin_amdgcn_global_load_async_to_lds_b128)
#    define SPOOL_ASYNC_LDS 1
#  endif
#endif
#ifndef SPOOL_ASYNC_LDS
#  define SPOOL_ASYNC_LDS 0
#endif

#define AS1 __attribute__((address_space(1)))
#define AS3 __attribute__((address_space(3)))

typedef int v4i __attribute__((ext_vector_type(4)));

__device__ __forceinline__ void copy16_g2lds(const float* g, float* l) {
#if SPOOL_ASYNC_LDS
    // Builtin signature (from hipcc diagnostic): params are pointers to int4
    // in global / LDS address spaces. Addrspace casts via integer round-trip
    // (generic LDS ptr low 32 bits == wave-relative LDS byte offset).
    AS1 v4i* gp = (AS1 v4i*)(unsigned long long)g;
    AS3 v4i* lp = (AS3 v4i*)(unsigned int)(unsigned long long)l;
    __builtin_amdgcn_global_load_async_to_lds_b128(gp, lp, 0, 0);
#else
    *(float4*)l = *(const float4*)g;
#endif
}

__device__ __forceinline__ void async_copies_done() {
#if SPOOL_ASYNC_LDS
#  if defined(__has_builtin) && __has_builtin(__builtin_amdgcn_s_wait_asynccnt)
    __builtin_amdgcn_s_wait_asynccnt(0);
#  else
    asm volatile("s_wait_asynccnt 0" ::: "memory");
#  endif
#endif
}

// Geometry constants (match reference)
#define IH   112
#define IW   112
#define OH   56
#define OW   56
#define TOH  4              // output rows per block
#define LR   9              // input rows per tile: 2*TOH+1
#define LW   120            // LDS row stride (floats); data cols at [4..115]

__global__ __launch_bounds__(256)
void SoftMaxPool2d_kernel(const float* __restrict__ in, float* __restrict__ out)
{
    __shared__ __align__(16) float smem[LR][LW];

    const int p    = blockIdx.y;          // plane index b*C + c
    const int oh0  = blockIdx.x * TOH;    // first output row of tile
    const int tid  = threadIdx.x;         // 0..255

    const float* plane = in + (size_t)p * (IH * IW);

    // LDS layout: padded col pc (0..113) lives at smem[pr][pc+3].
    // pc=0 is the left zero pad (offset 3); pc=113 (right pad) is never read
    // since max window col for ow=55 is pc=112 (offset 115).
    if (tid < LR) smem[tid][3] = 0.0f;    // left zero-pad column

    // Interior: 9 rows x 28 float4 (cols 0..111) = 252 16B transfers.
    if (tid < LR * 28) {
        const int pr = tid / 28;
        const int q  = tid - pr * 28;
        const int ih = 2 * oh0 - 1 + pr;  // global input row (pad=1)
        float* dst = &smem[pr][4 + 4 * q];            // 16B aligned
        if (ih >= 0 && ih < IH) {
            copy16_g2lds(plane + (size_t)ih * IW + 4 * q, dst);
        } else {
            float4 z = make_float4(0.f, 0.f, 0.f, 0.f);
            *(float4*)dst = z;                         // zero-pad row
        }
    }

    async_copies_done();
    __syncthreads();

    // Compute: thread (orow, ow) -> one output pixel.
    const int ow   = tid & 63;
    const int orow = tid >> 6;
    if (ow < OW) {
        // window rows in LDS: 2*orow + {0,1,2}; cols at offset 2*ow+3+{0,1,2}
        const float* r0 = &smem[2 * orow + 0][2 * ow + 3];
        const float* r1 = &smem[2 * orow + 1][2 * ow + 3];
        const float* r2 = &smem[2 * orow + 2][2 * ow + 3];

        float w[9];
        w[0] = r0[0]; w[1] = r0[1]; w[2] = r0[2];
        w[3] = r1[0]; w[4] = r1[1]; w[5] = r1[2];
        w[6] = r2[0]; w[7] = r2[1]; w[8] = r2[2];

        float m = w[0];
#pragma unroll
        for (int k = 1; k < 9; ++k) m = fmaxf(m, w[k]);

        float num = 0.f, den = 0.f;
#pragma unroll
        for (int k = 0; k < 9; ++k) {
            const float e = __expf(w[k] - m);   // T = 1.0
            den += e;
            num += e * w[k];
        }

        const int oh = oh0 + orow;
        out[((size_t)p * OH + oh) * OW + ow] = num / den;
    }
}

extern "C" void kernel_launch(void* const* d_in, const int* in_sizes, int n_in,
                              void* d_out, int out_size, void* d_ws, size_t ws_size,
                              hipStream_t stream)
{
    (void)n_in; (void)out_size; (void)d_ws; (void)ws_size;
    const float* in  = (const float*)d_in[0];
    float*       out = (float*)d_out;

    const int planes = in_sizes[0] / (IH * IW);   // 32*64 = 2048
    dim3 grid(OH / TOH, planes);                  // (14, 2048)
    SoftMaxPool2d_kernel<<<grid, 256, 0, stream>>>(in, out);
}